// MLP_44100724195774
// MI455X (gfx1250) — compile-verified
//
#include <hip/hip_runtime.h>
#include <hip/hip_bf16.h>
#include <math.h>

// Problem constants (match reference)
#define SEQ    3335
#define BATCH  4096
#define NPAIR  ((SEQ + 1) / 2)    // 1668 l-pairs (last one partial)
#define CHUNK  ((NPAIR + 7) / 8)  // 209 pairs per wave
#define NITER  ((CHUNK + 1) / 2)  // 105 two-pair iterations (covers 210 >= 209 pairs)
#define LCLAMP (SEQ - 1)

typedef __attribute__((ext_vector_type(2))) float v2f;
typedef __attribute__((ext_vector_type(8))) float v8f;

// D = A*B + C with V_WMMA_F32_16X16X4_F32 per l-pair:
//   A[m, 2j+c] = table[idx[row_m, l0+j]][c]  -> lane (16j+m)'s single b64 gather
//   B[2j+c, n] = (n == c) ? conv_w[c][l0+j] : 0
//   => D column 0 accumulates channel-0 sums, column 1 channel-1 sums.
// 8 waves/block split the pair range contiguously. Uniform padded trip count:
// invalid pairs are neutralized by clamped addresses + zeroed B (branchless).
__global__ __launch_bounds__(256) void MLP_44100724195774_kernel(
    const float* __restrict__ table,    // [VOCAB, 2]
    const float* __restrict__ conv_w,   // [2, SEQ]
    const float* __restrict__ dense_w,  // [2]
    const int*   __restrict__ idx,      // [BATCH, SEQ]
    float* __restrict__ out)            // [BATCH]
{
    __shared__ float lds_w[2 * SEQ];      // staged conv weights
    __shared__ float partial[8][2][16];   // [wave][channel][row-in-tile]

    const int tid  = threadIdx.x;
    const int wave = tid >> 5;
    const int lane = tid & 31;
    const int ncol = lane & 15;   // A row m / D column n
    const int half = lane >> 4;   // j: which element of the l-pair
    const int tile = blockIdx.x;
    const int row  = tile * 16 + ncol;

    // Stage conv_w into LDS (coalesced float2 loads; 2*SEQ == 6670 floats).
    {
        const v2f* cw2 = (const v2f*)conv_w;
        v2f* lw2 = (v2f*)lds_w;
        for (int j = tid; j < SEQ; j += 256) lw2[j] = cw2[j];
    }
    __syncthreads();

    const long long rowbase = (long long)row * SEQ;
    const int* __restrict__ iprow = idx + rowbase;
    const int wsel = (ncol == 1) ? SEQ : 0;       // which channel's weight row

    const int pbeg = wave * CHUNK;
    int lend = 2 * (pbeg + CHUNK);                // first invalid l for this wave
    if (lend > SEQ) lend = SEQ;                   // also kills the odd l = 3335

    // Clamped helpers: always in-bounds; invalid pairs contribute 0 via B.
    auto LDIDX = [&](int p) {
        int l = 2 * p + half;
        return iprow[l < LCLAMP ? l : LCLAMP];
    };
    auto GATH = [&](int t) { return *(const v2f*)(table + (long long)t * 2); };
    auto WLD = [&](int p) {
        int l = 2 * p + half;
        return lds_w[wsel + (l < LCLAMP ? l : LCLAMP)];
    };
    auto MKB = [&](float wv, int l) {
        const float wvv = (l < lend) ? wv : 0.0f;
        v2f b;
        b.x = (ncol == 0) ? wvv : 0.0f;
        b.y = (ncol == 1) ? wvv : 0.0f;
        return b;
    };

    // --- pipeline prologue (pairs pbeg .. pbeg+5) ---
    int p = pbeg;
    const int u0 = LDIDX(p);
    const int u1 = LDIDX(p + 1);
    int t0 = LDIDX(p + 2), t1 = LDIDX(p + 3);   // idx FIFO: ready slots
    int t2 = LDIDX(p + 4), t3 = LDIDX(p + 5);   // idx FIFO: in-flight slots
    v2f a0 = GATH(u0), a1 = GATH(u1);           // embeddings for p, p+1
    float wA = WLD(p), wB = WLD(p + 1);         // weights for p, p+1

    v8f acc0 = {}, acc1 = {};

    // --- steady state: uniform trip count, no guards, no epilogue ---
    for (int it = 0; it < NITER; ++it, p += 2) {
        v2f g0 = GATH(t0);                      // gathers for p+2, p+3
        v2f g1 = GATH(t1);
        const int n0 = LDIDX(p + 6);            // indices for p+6, p+7
        const int n1 = LDIDX(p + 7);
        __builtin_prefetch(iprow + 2 * p + 128, 0, 1);  // 512B ahead (speculative DEV)
        const float wn0 = WLD(p + 2);           // weights for p+2, p+3
        const float wn1 = WLD(p + 3);

        const v2f b0 = MKB(wA, 2 * p + half);
        const v2f b1 = MKB(wB, 2 * p + 2 + half);
        acc0 = __builtin_amdgcn_wmma_f32_16x16x4_f32(false, a0, false, b0,
                                                     (short)0, acc0, false, false);
        acc1 = __builtin_amdgcn_wmma_f32_16x16x4_f32(false, a1, false, b1,
                                                     (short)0, acc1, false, false);
        a0 = g0; a1 = g1;
        t0 = t2; t1 = t3; t2 = n0; t3 = n1;
        wA = wn0; wB = wn1;
    }

    v8f acc;
#pragma unroll
    for (int r = 0; r < 8; ++r) acc[r] = acc0[r] + acc1[r];

    // D layout: VGPR r, lanes 0-15 -> (M=r, N=lane), lanes 16-31 -> (M=8+r, N=lane-16).
    if (ncol < 2) {
        const int ch = ncol;
        const int mb = half * 8;
#pragma unroll
        for (int r = 0; r < 8; ++r) partial[wave][ch][mb + r] = acc[r];
    }
    __syncthreads();

    if (tid < 16) {
        float s0 = 0.0f, s1 = 0.0f;
#pragma unroll
        for (int w = 0; w < 8; ++w) {
            s0 += partial[w][0][tid];
            s1 += partial[w][1][tid];
        }
        // hardswish(x) = x * clamp(x+3, 0, 6) / 6
        const float c0 = fminf(fmaxf(s0 + 3.0f, 0.0f), 6.0f);
        const float c1 = fminf(fmaxf(s1 + 3.0f, 0.0f), 6.0f);
        const float h0 = s0 * c0 * (1.0f / 6.0f);
        const float h1 = s1 * c1 * (1.0f / 6.0f);
        const float y  = h0 * dense_w[0] + h1 * dense_w[1];
        out[tile * 16 + tid] = tanhf(y);
    }
}

extern "C" void kernel_launch(void* const* d_in, const int* in_sizes, int n_in,
                              void* d_out, int out_size, void* d_ws, size_t ws_size,
                              hipStream_t stream) {
    const float* table   = (const float*)d_in[0];   // [VOCAB, 2] f32
    const float* conv_w  = (const float*)d_in[1];   // [2, SEQ]  f32
    const float* dense_w = (const float*)d_in[2];   // [2]       f32
    const int*   inputs  = (const int*)d_in[3];     // [BATCH, SEQ] int32
    float* out = (float*)d_out;                     // [BATCH] f32

    dim3 grid(BATCH / 16);   // 256 tiles of 16 batch rows
    dim3 block(256);         // 8 waves per tile, contiguous l-pair chunks
    hipLaunchKernelGGL(MLP_44100724195774_kernel, grid, block, 0, stream,
                       table, conv_w, dense_w, inputs, out);
}